// OCNN_LeNet_8624294330838
// MI455X (gfx1250) — compile-verified
//
#include <hip/hip_runtime.h>

// OCNN-LeNet forward for MI455X (gfx1250, wave32).
// Conv stages: 27-neighbor gather -> fp32 WMMA (V_WMMA_F32_16X16X4_F32),
// BN+ReLU+octree-max-pool fused in registers per 16-node tile.
// Weight B-fragments pre-swizzled into LDS (ds_load_b64 in the hot loop);
// gather uses unconditional clamped global_load_b64 + cndmask (no EXEC branches).

typedef __attribute__((ext_vector_type(2))) float v2f;
typedef __attribute__((ext_vector_type(8))) float v8f;

#define OC_N5 1048576
#define OC_N4 131072
#define OC_N3 16384
#define OC_B  32
#define OC_EPS 1e-5f

__device__ __forceinline__ v8f wmma_f32_k4(v2f a, v2f b, v8f c) {
  // D(16x16,f32) = A(16x4,f32) * B(4x16,f32) + C
  return __builtin_amdgcn_wmma_f32_16x16x4_f32(false, a, false, b,
                                               (short)0, c, false, false);
}

// ---------------- Stage 1: Cin=4 -> Cout=16, pool 8:1 ----------------
// LDS: 27 B-fragments * 32 lanes * 8B = 6912 B
__global__ void __launch_bounds__(256) oc_stage1(
    const float* __restrict__ feat, const int* __restrict__ neigh,
    const float* __restrict__ w,      // [27,4,16]
    const float* __restrict__ gamma, const float* __restrict__ beta,
    float* __restrict__ out)          // [N4,16]
{
  extern __shared__ float2 wl[];
  // Pre-swizzle B fragments: wl[k*32+l] = { w[k][2h][r], w[k][2h+1][r] }
  for (int e = threadIdx.x; e < 27 * 32; e += 256) {
    const int k = e >> 5, l = e & 31;
    const int r = l & 15, h = l >> 4;
    float2 t;
    t.x = w[(k * 4 + h * 2 + 0) * 16 + r];
    t.y = w[(k * 4 + h * 2 + 1) * 16 + r];
    wl[e] = t;
  }
  __syncthreads();

  const int wave = blockIdx.x * 8 + (threadIdx.x >> 5);
  const int lane = threadIdx.x & 31;
  const int row  = lane & 15;
  const int half = lane >> 4;
  const int node = wave * 16 + row;
  const int* nrow = neigh + (size_t)node * 27;

  v8f acc0 = {}, acc1 = {};
#pragma unroll            // full unroll: parity below is compile-time constant
  for (int k = 0; k < 27; ++k) {
    const int idx = nrow[k];
    const bool valid = idx >= 0;
    const unsigned uidx = valid ? (unsigned)idx : 0u;   // zero-extend, no ashr
    const float2 av = *(const float2*)(feat + (size_t)uidx * 4 + half * 2);
    v2f a;
    a.x = valid ? av.x : 0.0f;
    a.y = valid ? av.y : 0.0f;
    const float2 bv = wl[k * 32 + lane];
    v2f b; b.x = bv.x; b.y = bv.y;
    if (k & 1) acc1 = wmma_f32_k4(a, b, acc1);
    else       acc0 = wmma_f32_k4(a, b, acc0);
  }

  const float sc = gamma[row] * rsqrtf(1.0f + OC_EPS);
  const float bt = beta[row];
  float pm = 0.0f;                  // ReLU >= 0, so 0-init is exact for the max
#pragma unroll
  for (int r = 0; r < 8; ++r)
    pm = fmaxf(pm, fmaxf((acc0[r] + acc1[r]) * sc + bt, 0.0f));
  out[(size_t)(wave * 2 + half) * 16 + row] = pm;
}

// ---------------- Stage 2: Cin=16 -> Cout=32, pool 8:1 ----------------
// LDS: 27*4*2 fragments * 32 * 8B = 55296 B
__global__ void __launch_bounds__(256) oc_stage2(
    const float* __restrict__ x, const int* __restrict__ neigh,
    const float* __restrict__ w,      // [27,16,32]
    const float* __restrict__ gamma, const float* __restrict__ beta,
    float* __restrict__ out)          // [N3,32]
{
  extern __shared__ float2 wl[];
  for (int e = threadIdx.x; e < 27 * 4 * 2 * 32; e += 256) {
    const int l = e & 31;
    const int frag = e >> 5;              // ((k*4 + cb)*2 + t)
    const int t  = frag & 1;
    const int cb = (frag >> 1) & 3;
    const int k  = frag >> 3;
    const int r = l & 15, h = l >> 4;
    const float* wp = w + (size_t)(k * 16 + cb * 4 + h * 2) * 32 + t * 16 + r;
    float2 v; v.x = wp[0]; v.y = wp[32];
    wl[e] = v;
  }
  __syncthreads();

  const int wave = blockIdx.x * 8 + (threadIdx.x >> 5);
  const int lane = threadIdx.x & 31;
  const int row  = lane & 15;
  const int half = lane >> 4;
  const int node = wave * 16 + row;
  const int* nrow = neigh + (size_t)node * 27;

  v8f acc0 = {}, acc1 = {};
#pragma unroll 1
  for (int k = 0; k < 27; ++k) {
    const int idx = nrow[k];
    const bool valid = idx >= 0;
    const unsigned uidx = valid ? (unsigned)idx : 0u;
    const float* px = x + (size_t)uidx * 16 + half * 2;
#pragma unroll
    for (int cb = 0; cb < 4; ++cb) {
      const float2 av = *(const float2*)(px + cb * 4);
      v2f a;
      a.x = valid ? av.x : 0.0f;
      a.y = valid ? av.y : 0.0f;
      const float2 b0v = wl[((k * 4 + cb) * 2 + 0) * 32 + lane];
      const float2 b1v = wl[((k * 4 + cb) * 2 + 1) * 32 + lane];
      v2f b0; b0.x = b0v.x; b0.y = b0v.y;
      v2f b1; b1.x = b1v.x; b1.y = b1v.y;
      acc0 = wmma_f32_k4(a, b0, acc0);
      acc1 = wmma_f32_k4(a, b1, acc1);
    }
  }

  const float rs = rsqrtf(1.0f + OC_EPS);
  const float sc0 = gamma[row] * rs,      bt0 = beta[row];
  const float sc1 = gamma[16 + row] * rs, bt1 = beta[16 + row];
  float p0 = 0.0f, p1 = 0.0f;
#pragma unroll
  for (int r = 0; r < 8; ++r) {
    p0 = fmaxf(p0, fmaxf(acc0[r] * sc0 + bt0, 0.0f));
    p1 = fmaxf(p1, fmaxf(acc1[r] * sc1 + bt1, 0.0f));
  }
  const size_t parent = (size_t)(wave * 2 + half);
  out[parent * 32 + row]      = p0;
  out[parent * 32 + 16 + row] = p1;
}

// ---------------- Stage 3: Cin=32 -> Cout=64, pool 8:1 ----------------
// LDS: 27*8*4 fragments * 32 * 8B = 221184 B (fits 320KB WGP LDS)
__global__ void __launch_bounds__(256) oc_stage3(
    const float* __restrict__ x, const int* __restrict__ neigh,
    const float* __restrict__ w,      // [27,32,64]
    const float* __restrict__ gamma, const float* __restrict__ beta,
    float* __restrict__ out)          // [2048,64]
{
  extern __shared__ float2 wl[];
  for (int e = threadIdx.x; e < 27 * 8 * 4 * 32; e += 256) {
    const int l = e & 31;
    const int frag = e >> 5;              // ((k*8 + cb)*4 + t)
    const int t  = frag & 3;
    const int cb = (frag >> 2) & 7;
    const int k  = frag >> 5;
    const int r = l & 15, h = l >> 4;
    const float* wp = w + (size_t)(k * 32 + cb * 4 + h * 2) * 64 + t * 16 + r;
    float2 v; v.x = wp[0]; v.y = wp[64];
    wl[e] = v;
  }
  __syncthreads();

  const int wave = blockIdx.x * 8 + (threadIdx.x >> 5);
  const int lane = threadIdx.x & 31;
  const int row  = lane & 15;
  const int half = lane >> 4;
  const int node = wave * 16 + row;
  const int* nrow = neigh + (size_t)node * 27;

  v8f acc[4] = {{}, {}, {}, {}};
#pragma unroll 1
  for (int k = 0; k < 27; ++k) {
    const int idx = nrow[k];
    const bool valid = idx >= 0;
    const unsigned uidx = valid ? (unsigned)idx : 0u;
    const float* px = x + (size_t)uidx * 32 + half * 2;
#pragma unroll
    for (int cb = 0; cb < 8; ++cb) {
      const float2 av = *(const float2*)(px + cb * 4);
      v2f a;
      a.x = valid ? av.x : 0.0f;
      a.y = valid ? av.y : 0.0f;
#pragma unroll
      for (int t = 0; t < 4; ++t) {
        const float2 bv = wl[((k * 8 + cb) * 4 + t) * 32 + lane];
        v2f b; b.x = bv.x; b.y = bv.y;
        acc[t] = wmma_f32_k4(a, b, acc[t]);
      }
    }
  }

  const float rs = rsqrtf(1.0f + OC_EPS);
  const size_t parent = (size_t)(wave * 2 + half);
#pragma unroll
  for (int t = 0; t < 4; ++t) {
    const int ch = t * 16 + row;
    const float sc = gamma[ch] * rs, bt = beta[ch];
    float pm = 0.0f;
#pragma unroll
    for (int r = 0; r < 8; ++r)
      pm = fmaxf(pm, fmaxf(acc[t][r] * sc + bt, 0.0f));
    out[parent * 64 + ch] = pm;
  }
}

// ---------------- FC1: voxelize + [32,4096]x[4096,128] + BN + ReLU ----------------
// One block per batch; voxel vector staged in LDS; lanes coalesce over 128 cols.
__global__ void __launch_bounds__(128) oc_fc1(
    const float* __restrict__ x3,     // [B*64 nodes, 64 ch]
    const float* __restrict__ fw,     // [4096,128]
    const float* __restrict__ hg, const float* __restrict__ hb,
    float* __restrict__ h)            // [B,128]
{
  __shared__ float v[4096];
  const int b = blockIdx.x, o = threadIdx.x;
  for (int j = o; j < 4096; j += 128) {
    const int ch = j >> 6, nd = j & 63;   // v[ch*64+nd] = x3[b*64+nd, ch]
    v[j] = x3[(size_t)(b * 64 + nd) * 64 + ch];
  }
  __syncthreads();
  float s = 0.0f;
  const float* wp = fw + o;
#pragma unroll 4
  for (int j = 0; j < 4096; ++j) s += v[j] * wp[(size_t)j * 128];
  const float r = s * hg[o] * rsqrtf(1.0f + OC_EPS) + hb[o];
  h[b * 128 + o] = fmaxf(r, 0.0f);
}

// ---------------- FC2 (128->64) + regressor (64->1) ----------------
__global__ void __launch_bounds__(64) oc_head(
    const float* __restrict__ h, const float* __restrict__ w2,
    const float* __restrict__ b2, const float* __restrict__ rw,
    const float* __restrict__ rb, float* __restrict__ out)
{
  __shared__ float zs[64];
  const int b = blockIdx.x, o = threadIdx.x;
  float s = b2[o];
  for (int k = 0; k < 128; ++k) s += h[b * 128 + k] * w2[k * 64 + o];
  zs[o] = s;
  __syncthreads();
  if (o == 0) {
    float r = rb[0];
    for (int i = 0; i < 64; ++i) r += zs[i] * rw[i];
    out[b] = r;
  }
}

extern "C" void kernel_launch(void* const* d_in, const int* in_sizes, int n_in,
                              void* d_out, int out_size, void* d_ws, size_t ws_size,
                              hipStream_t stream) {
  (void)in_sizes; (void)n_in; (void)out_size; (void)ws_size;
  const float* feat = (const float*)d_in[0];
  const float* w1   = (const float*)d_in[1];
  const float* g1   = (const float*)d_in[2];
  const float* b1   = (const float*)d_in[3];
  const float* w2   = (const float*)d_in[4];
  const float* g2   = (const float*)d_in[5];
  const float* b2   = (const float*)d_in[6];
  const float* w3   = (const float*)d_in[7];
  const float* g3   = (const float*)d_in[8];
  const float* b3   = (const float*)d_in[9];
  const float* fc1w = (const float*)d_in[10];
  const float* hg   = (const float*)d_in[11];
  const float* hb   = (const float*)d_in[12];
  const float* fc2w = (const float*)d_in[13];
  const float* fc2b = (const float*)d_in[14];
  const float* regw = (const float*)d_in[15];
  const float* regb = (const float*)d_in[16];
  const int* neigh5 = (const int*)d_in[17];
  const int* neigh4 = (const int*)d_in[18];
  const int* neigh3 = (const int*)d_in[19];

  float* x1 = (float*)d_ws;                 // [N4,16]   8 MB
  float* x2 = x1 + (size_t)OC_N4 * 16;      // [N3,32]   2 MB
  float* x3 = x2 + (size_t)OC_N3 * 32;      // [2048,64] 0.5 MB
  float* h  = x3 + (size_t)2048 * 64;       // [B,128]   16 KB
  float* out = (float*)d_out;

  oc_stage1<<<OC_N5 / 16 / 8, 256, 27 * 32 * sizeof(float2), stream>>>(
      feat, neigh5, w1, g1, b1, x1);
  oc_stage2<<<OC_N4 / 16 / 8, 256, 27 * 4 * 2 * 32 * sizeof(float2), stream>>>(
      x1, neigh4, w2, g2, b2, x2);
  oc_stage3<<<OC_N3 / 16 / 8, 256, 27 * 8 * 4 * 32 * sizeof(float2), stream>>>(
      x2, neigh3, w3, g3, b3, x3);
  oc_fc1<<<OC_B, 128, 0, stream>>>(x3, fc1w, hg, hb, h);
  oc_head<<<OC_B, 64, 0, stream>>>(h, fc2w, fc2b, regw, regb, out);
}